// DecoderCategorical_55336358642820
// MI455X (gfx1250) — compile-verified
//
#include <hip/hip_runtime.h>
#include <hip/hip_bf16.h>
#include <stdint.h>

typedef __attribute__((ext_vector_type(2))) float v2f;
typedef __attribute__((ext_vector_type(8))) float v8f;

#define GDIM   25000
#define BDIM   256
#define HDIM   128
#define CDIM   7
#define GBDIM  1000
#define BINSZ  25
#define NHALF  22400000u   // (B*G*C)/2 == 128*G*C : JAX threefry counter pairing

// ---------------- threefry2x32 (JAX-compatible: key = (0, 42)) ----------------
__device__ __forceinline__ uint32_t rotl32(uint32_t x, int r) {
    return (x << r) | (x >> (32 - r));
}

__device__ __forceinline__ void threefry2x32(uint32_t x0, uint32_t x1,
                                             uint32_t& o0, uint32_t& o1) {
    const uint32_t k0 = 0u, k1 = 42u;
    const uint32_t k2 = k0 ^ k1 ^ 0x1BD11BDAu;
    const uint32_t ks[3] = {k0, k1, k2};
    const int RA[4] = {13, 15, 26, 6};
    const int RB[4] = {17, 29, 16, 24};
    x0 += ks[0];
    x1 += ks[1];
#pragma unroll
    for (int grp = 0; grp < 5; ++grp) {
        const int* R = (grp & 1) ? RB : RA;
#pragma unroll
        for (int j = 0; j < 4; ++j) {
            x0 += x1;
            x1 = rotl32(x1, R[j]);
            x1 ^= x0;
        }
        x0 += ks[(grp + 1) % 3];
        x1 += ks[(grp + 2) % 3] + (uint32_t)(grp + 1);
    }
    o0 = x0;
    o1 = x1;
}

__device__ __forceinline__ float bits_to_uniform(uint32_t w) {
    uint32_t bits = (w >> 9) | 0x3F800000u;  // [1,2)
    float f;
    __builtin_memcpy(&f, &bits, 4);
    return f - 1.0f;                         // [0,1)
}

// ---------------- tiny MLP: Dense -> BN -> ReLU (x2) ----------------
__global__ void mlp_layer0(const float* __restrict__ z, const float* __restrict__ w0,
                           const float* __restrict__ b0, const float* __restrict__ g0,
                           const float* __restrict__ be0, const float* __restrict__ m0,
                           const float* __restrict__ v0, float* __restrict__ x0) {
    int idx = blockIdx.x * blockDim.x + threadIdx.x;  // 256*128
    int b = idx >> 7, h = idx & 127;
    float acc = b0[h];
#pragma unroll
    for (int k = 0; k < 20; ++k) acc += z[b * 20 + k] * w0[k * HDIM + h];
    float t = (acc - m0[h]) * rsqrtf(v0[h] + 1e-3f) * g0[h] + be0[h];
    x0[idx] = fmaxf(t, 0.0f);
}

__global__ void mlp_layer1(const float* __restrict__ x0, const float* __restrict__ w1,
                           const float* __restrict__ b1, const float* __restrict__ g1,
                           const float* __restrict__ be1, const float* __restrict__ m1,
                           const float* __restrict__ v1, float* __restrict__ x) {
    int idx = blockIdx.x * blockDim.x + threadIdx.x;
    int b = idx >> 7, h = idx & 127;
    float acc = b1[h];
#pragma unroll 8
    for (int k = 0; k < HDIM; ++k) acc += x0[b * HDIM + k] * w1[k * HDIM + h];
    float t = (acc - m1[h]) * rsqrtf(v1[h] + 1e-3f) * g1[h] + be1[h];
    x[idx] = fmaxf(t, 0.0f);
}

// ---------------- rho heads: 7x GEMM [256x128]x[128x1000] via WMMA f32 ----------------
// logits layout: [B][C][GB]
__global__ __launch_bounds__(256) void rho_gemm(const float* __restrict__ x,
                                                const float* __restrict__ wrho,
                                                const float* __restrict__ brho,
                                                float* __restrict__ logits) {
    int wave = (blockIdx.x * blockDim.x + threadIdx.x) >> 5;  // 16 * 63 * 7 waves
    int lane = threadIdx.x & 31;
    int tm = wave & 15;
    int rest = wave >> 4;          // 0..440
    int tnb = rest % 63;           // GB tile
    int c = rest / 63;             // copy-number head
    int lmod = lane & 15, lhalf = lane >> 4;

    int arow = tm * 16 + lmod;
    int col = tnb * 16 + lmod;
    int colc = (col < GBDIM) ? col : (GBDIM - 1);
    bool cval = (col < GBDIM);

    const float* w = wrho + (size_t)c * HDIM * GBDIM;
    const float* xrow = x + arow * HDIM + lhalf * 2;  // A frag: K=koff,koff+1
    v8f acc = {};
#pragma unroll 4
    for (int k = 0; k < HDIM; k += 4) {
        v2f a;
        a.x = xrow[k];
        a.y = xrow[k + 1];
        int r0 = (k + lhalf * 2) * GBDIM + colc;
        v2f bf;
        bf.x = w[r0];
        bf.y = w[r0 + GBDIM];
        if (!cval) { bf.x = 0.0f; bf.y = 0.0f; }
        acc = __builtin_amdgcn_wmma_f32_16x16x4_f32(false, a, false, bf, (short)0, acc,
                                                    false, false);
    }
    if (!cval) return;
    float bb = brho[c * GBDIM + col];
#pragma unroll
    for (int i = 0; i < 8; ++i) {
        int brow = tm * 16 + lhalf * 8 + i;
        logits[(size_t)brow * (CDIM * GBDIM) + c * GBDIM + col] = acc[i] + bb;
    }
}

// softmax over C, re-layout rho -> [B][GB][C] (7 contiguous per (b,gb))
__global__ void rho_softmax(const float* __restrict__ logits, float* __restrict__ rho) {
    int idx = blockIdx.x * blockDim.x + threadIdx.x;  // 256*1000
    if (idx >= BDIM * GBDIM) return;
    int b = idx / GBDIM, gb = idx % GBDIM;
    float l[CDIM];
    float mx = -1e30f;
#pragma unroll
    for (int c = 0; c < CDIM; ++c) {
        l[c] = logits[(size_t)b * (CDIM * GBDIM) + c * GBDIM + gb];
        mx = fmaxf(mx, l[c]);
    }
    float s = 0.0f;
#pragma unroll
    for (int c = 0; c < CDIM; ++c) { l[c] = __expf(l[c] - mx); s += l[c]; }
    float inv = 1.0f / s;
#pragma unroll
    for (int c = 0; c < CDIM; ++c)
        rho[(size_t)b * (GBDIM * CDIM) + gb * CDIM + c] = l[c] * inv;
}

// ---------------- fused GEMM epilogue: theta = exp(x@wr+br), pi_drop = x@wd+bd ----------------
__global__ __launch_bounds__(256) void fused_main(
    const float* __restrict__ x, const float* __restrict__ wr, const float* __restrict__ br,
    const float* __restrict__ wd, const float* __restrict__ bd,
    float* __restrict__ theta, float* __restrict__ pid) {
    int wave = (blockIdx.x * blockDim.x + threadIdx.x) >> 5;  // 16 * 1563 waves
    int lane = threadIdx.x & 31;
    int tm = wave & 15;
    int tn = wave >> 4;  // 0..1562
    int lmod = lane & 15, lhalf = lane >> 4;

    int arow = tm * 16 + lmod;
    int col = tn * 16 + lmod;
    int colc = (col < GDIM) ? col : (GDIM - 1);
    bool cval = (col < GDIM);

    const float* xrow = x + arow * HDIM + lhalf * 2;
    v8f cr = {};
    v8f cd = {};
#pragma unroll 4
    for (int k = 0; k < HDIM; k += 4) {
        v2f a;
        a.x = xrow[k];
        a.y = xrow[k + 1];
        int r0 = (k + lhalf * 2) * GDIM + colc;
        v2f b1, b2;
        b1.x = wr[r0];
        b1.y = wr[r0 + GDIM];
        b2.x = wd[r0];
        b2.y = wd[r0 + GDIM];
        if (!cval) { b1.x = 0.f; b1.y = 0.f; b2.x = 0.f; b2.y = 0.f; }
        cr = __builtin_amdgcn_wmma_f32_16x16x4_f32(false, a, false, b1, (short)0, cr,
                                                   false, false);
        cd = __builtin_amdgcn_wmma_f32_16x16x4_f32(false, a, false, b2, (short)0, cd,
                                                   false, false);
    }
    if (!cval) return;  // divergence only after all WMMAs

    float brn = br[col], bdn = bd[col];
#pragma unroll
    for (int i = 0; i < 8; ++i) {
        int brow = tm * 16 + lhalf * 8 + i;
        size_t off = (size_t)brow * GDIM + col;
        theta[off] = __expf(cr[i] + brn);
        pid[off] = cd[i] + bdn;
    }
}

// ---------------- gumbel argmax sample + sigmoid scale ----------------
// JAX pairing: uniform[(b,g,c)] and uniform[(b+128,g,c)] come from ONE threefry call
// on counters (j, j+N/2) with j = (b*G+g)*C + c  (b < 128). 3.5 calls per element.
__global__ void gumbel_sample(const float* __restrict__ rho,
                              const float* __restrict__ wsc, const float* __restrict__ bsc,
                              float* __restrict__ mu, float* __restrict__ samp) {
    int idx = blockIdx.x * blockDim.x + threadIdx.x;  // 128 * 25000
    int b = idx / GDIM;                               // 0..127
    int g = idx - b * GDIM;
    int gb = g / BINSZ;

    const float* rlo = rho + (size_t)b * (GBDIM * CDIM) + gb * CDIM;
    const float* rhi = rlo + (size_t)128 * (GBDIM * CDIM);
    uint32_t j0 = ((uint32_t)b * (uint32_t)GDIM + (uint32_t)g) * (uint32_t)CDIM;

    float bestLo = -1e30f, bestHi = -1e30f;
    int biLo = 0, biHi = 0;
#pragma unroll
    for (int c = 0; c < CDIM; ++c) {
        uint32_t o0, o1;
        threefry2x32(j0 + (uint32_t)c, j0 + (uint32_t)c + NHALF, o0, o1);
        float ulo = bits_to_uniform(o0);
        float uhi = bits_to_uniform(o1);
        float glo = -__logf(-__logf(ulo + 1e-20f) + 1e-20f);
        float ghi = -__logf(-__logf(uhi + 1e-20f) + 1e-20f);
        float slo = __logf(rlo[c] + 1e-20f) + glo;  // /TEMP + softmax are monotonic
        float shi = __logf(rhi[c] + 1e-20f) + ghi;
        if (slo > bestLo) { bestLo = slo; biLo = c; }
        if (shi > bestHi) { bestHi = shi; biHi = c; }
    }
    float wsn = wsc[g], bsn = bsc[g];
    size_t off = (size_t)b * GDIM + g;
    float sLo = (float)biLo;   // straight-through forward == one-hot argmax
    float sHi = (float)biHi;
    samp[off] = sLo;
    mu[off] = 1.0f / (1.0f + __expf(-(sLo * wsn + bsn)));
    size_t off2 = off + (size_t)128 * GDIM;
    samp[off2] = sHi;
    mu[off2] = 1.0f / (1.0f + __expf(-(sHi * wsn + bsn)));
}

// ---------------- pi broadcast: [B][GB][7] -> [B][G][7], b128 stores ----------------
__global__ void pi_bcast(const float* __restrict__ rho, float* __restrict__ pi) {
    unsigned int t = blockIdx.x * blockDim.x + threadIdx.x;  // (B*G*C)/4 threads
    unsigned int o = t * 4u;                                 // flat into B*G*C
    unsigned int b = o / (GDIM * CDIM);                      // G*C % 4 == 0: same b for all 4
    unsigned int rem = o - b * (GDIM * CDIM);
    const float* rb = rho + (size_t)b * (GBDIM * CDIM);
    float4 v;
    float* pv = &v.x;
#pragma unroll
    for (int kk = 0; kk < 4; ++kk) {
        unsigned int r2 = rem + kk;
        unsigned int g = r2 / CDIM;
        unsigned int c = r2 - g * CDIM;
        pv[kk] = rb[(g / BINSZ) * CDIM + c];
    }
    *(float4*)(pi + o) = v;
}

extern "C" void kernel_launch(void* const* d_in, const int* in_sizes, int n_in,
                              void* d_out, int out_size, void* d_ws, size_t ws_size,
                              hipStream_t stream) {
    const float* z   = (const float*)d_in[0];
    const float* w0  = (const float*)d_in[1];
    const float* b0  = (const float*)d_in[2];
    const float* g0  = (const float*)d_in[3];
    const float* be0 = (const float*)d_in[4];
    const float* m0  = (const float*)d_in[5];
    const float* v0  = (const float*)d_in[6];
    const float* w1  = (const float*)d_in[7];
    const float* b1  = (const float*)d_in[8];
    const float* g1  = (const float*)d_in[9];
    const float* be1 = (const float*)d_in[10];
    const float* m1  = (const float*)d_in[11];
    const float* v1  = (const float*)d_in[12];
    const float* wr  = (const float*)d_in[13];
    const float* br  = (const float*)d_in[14];
    const float* wd  = (const float*)d_in[15];
    const float* bd  = (const float*)d_in[16];
    const float* wrho = (const float*)d_in[17];
    const float* brho = (const float*)d_in[18];
    const float* wsc = (const float*)d_in[19];
    const float* bsc = (const float*)d_in[20];

    float* ws = (float*)d_ws;
    float* x0     = ws;                       // 256*128
    float* x      = ws + 32768;               // 256*128
    float* logits = ws + 65536;               // 256*7000, layout [B][C][GB]
    float* rho    = ws + 65536 + 1792000;     // 256*7000, layout [B][GB][7]

    float* out    = (float*)d_out;
    float* mu     = out;                            // B*G
    float* theta  = out + (size_t)BDIM * GDIM;      // B*G
    float* pid    = out + 2 * (size_t)BDIM * GDIM;  // B*G
    float* samp   = out + 3 * (size_t)BDIM * GDIM;  // B*G
    float* pi     = out + 4 * (size_t)BDIM * GDIM;  // B*G*C

    // MLP trunk (tiny)
    mlp_layer0<<<128, 256, 0, stream>>>(z, w0, b0, g0, be0, m0, v0, x0);
    mlp_layer1<<<128, 256, 0, stream>>>(x0, w1, b1, g1, be1, m1, v1, x);

    // rho heads: 16 M-tiles * 63 N-tiles * 7 heads = 7056 waves / 8 per block
    rho_gemm<<<882, 256, 0, stream>>>(x, wrho, brho, logits);
    rho_softmax<<<1000, 256, 0, stream>>>(logits, rho);

    // sample + mu: 128*25000 threads, each produces rows b and b+128 (paired RNG)
    gumbel_sample<<<12500, 256, 0, stream>>>(rho, wsc, bsc, mu, samp);

    // theta / pi_drop WMMA GEMMs: 16 * 1563 = 25008 waves / 8 per block
    fused_main<<<3126, 256, 0, stream>>>(x, wr, br, wd, bd, theta, pid);

    // pi broadcast: (256*25000*7)/4 float4 stores
    pi_bcast<<<43750, 256, 0, stream>>>(rho, pi);
}